// MonarchMatrix_24558622998772
// MI455X (gfx1250) — compile-verified
//
#include <hip/hip_runtime.h>

typedef __attribute__((ext_vector_type(2))) float v2f;
typedef __attribute__((ext_vector_type(8))) float v8f;

#define S       64          // monarch block size (sqrt N)
#define N_FULL  4096        // S*S
#define TB      16          // batch rows per workgroup (WMMA M)
#define CS      65          // LDS stride of c-dim (b padded 64->65)
#define SR      4164        // LDS stride of r-dim (64*65 + 4; ==4 mod 64 -> no bank conflicts)
#define NWAVES  8
#define NT      (NWAVES * 32)

__global__ __launch_bounds__(NT)
void monarch_wmma_f32_kernel(const float* __restrict__ x,
                             const float* __restrict__ L,
                             const float* __restrict__ R,
                             float* __restrict__ out)
{
    // Tt[r][c][b] at offset r*SR + c*CS + b   (dwords), ~260 KB
    __shared__ float Tt[TB * SR];

    const int lane = threadIdx.x & 31;
    const int wave = threadIdx.x >> 5;
    const int r0   = blockIdx.x * TB;

    const int lm = lane & 15;   // lane % 16 -> M (A) / N (B,D)
    const int lh = lane >> 4;   // lane / 16
    const int kb = 2 * lh;      // per-lane K base within a 4-wide K step

    const v8f vzero = {0.f, 0.f, 0.f, 0.f, 0.f, 0.f, 0.f, 0.f};

    // ----------------- Stage 1 -----------------
    // T[b][c] = sum_m L[b][c][m] * x[r][m*64 + b]
    // A[M=r'][K=m] = x[(r0+r')*4096 + m*64 + b]
    // B[K=m][N=c'] = L[b*4096 + c'*64 + m]
    const float* xrow = x + (size_t)(r0 + lm) * N_FULL;
    for (int b = wave; b < S; b += NWAVES) {
        v8f acc0 = vzero, acc1 = vzero, acc2 = vzero, acc3 = vzero;
        const float* Lb = L + b * (S * S);
        for (int k0 = 0; k0 < S; k0 += 4) {
            const int m = k0 + kb;
            v2f a;
            a.x = xrow[m * S + b];
            a.y = xrow[(m + 1) * S + b];

            v2f b0, b1, b2, b3;
            b0.x = Lb[( 0 + lm) * S + m];  b0.y = Lb[( 0 + lm) * S + m + 1];
            b1.x = Lb[(16 + lm) * S + m];  b1.y = Lb[(16 + lm) * S + m + 1];
            b2.x = Lb[(32 + lm) * S + m];  b2.y = Lb[(32 + lm) * S + m + 1];
            b3.x = Lb[(48 + lm) * S + m];  b3.y = Lb[(48 + lm) * S + m + 1];

            acc0 = __builtin_amdgcn_wmma_f32_16x16x4_f32(false, a, false, b0, (short)0, acc0, false, false);
            acc1 = __builtin_amdgcn_wmma_f32_16x16x4_f32(false, a, false, b1, (short)0, acc1, false, false);
            acc2 = __builtin_amdgcn_wmma_f32_16x16x4_f32(false, a, false, b2, (short)0, acc2, false, false);
            acc3 = __builtin_amdgcn_wmma_f32_16x16x4_f32(false, a, false, b3, (short)0, acc3, false, false);
        }
        // D element (M = v + 8*lh, N = t*16 + lm) -> Tt[M][N][b]
        #pragma unroll
        for (int v = 0; v < 8; ++v) {
            const int M = v + 8 * lh;
            Tt[M * SR + ( 0 + lm) * CS + b] = acc0[v];
            Tt[M * SR + (16 + lm) * CS + b] = acc1[v];
            Tt[M * SR + (32 + lm) * CS + b] = acc2[v];
            Tt[M * SR + (48 + lm) * CS + b] = acc3[v];
        }
    }

    __syncthreads();

    // ----------------- Stage 2 -----------------
    // out[r][n*64 + c] = sum_b R[c][n][b] * T[b][c]
    // A[M=r'][K=b] = Tt[r'][c][b]   (consecutive in b -> ds b64 loads)
    // B[K=b][N=n'] = R[c*4096 + n'*64 + b]
    for (int c = wave; c < S; c += NWAVES) {
        v8f acc0 = vzero, acc1 = vzero, acc2 = vzero, acc3 = vzero;
        const float* Rc = R + c * (S * S);
        const float* Ta = &Tt[lm * SR + c * CS];
        for (int k0 = 0; k0 < S; k0 += 4) {
            const int bi = k0 + kb;
            v2f a;
            a.x = Ta[bi];
            a.y = Ta[bi + 1];

            v2f b0, b1, b2, b3;
            b0.x = Rc[( 0 + lm) * S + bi];  b0.y = Rc[( 0 + lm) * S + bi + 1];
            b1.x = Rc[(16 + lm) * S + bi];  b1.y = Rc[(16 + lm) * S + bi + 1];
            b2.x = Rc[(32 + lm) * S + bi];  b2.y = Rc[(32 + lm) * S + bi + 1];
            b3.x = Rc[(48 + lm) * S + bi];  b3.y = Rc[(48 + lm) * S + bi + 1];

            acc0 = __builtin_amdgcn_wmma_f32_16x16x4_f32(false, a, false, b0, (short)0, acc0, false, false);
            acc1 = __builtin_amdgcn_wmma_f32_16x16x4_f32(false, a, false, b1, (short)0, acc1, false, false);
            acc2 = __builtin_amdgcn_wmma_f32_16x16x4_f32(false, a, false, b2, (short)0, acc2, false, false);
            acc3 = __builtin_amdgcn_wmma_f32_16x16x4_f32(false, a, false, b3, (short)0, acc3, false, false);
        }
        // D element (M = v + 8*lh, N = t*16 + lm) -> out[(r0+M)*4096 + N*64 + c]
        #pragma unroll
        for (int v = 0; v < 8; ++v) {
            const int M = v + 8 * lh;
            float* orow = out + (size_t)(r0 + M) * N_FULL + c;
            orow[( 0 + lm) * S] = acc0[v];
            orow[(16 + lm) * S] = acc1[v];
            orow[(32 + lm) * S] = acc2[v];
            orow[(48 + lm) * S] = acc3[v];
        }
    }
}

extern "C" void kernel_launch(void* const* d_in, const int* in_sizes, int n_in,
                              void* d_out, int out_size, void* d_ws, size_t ws_size,
                              hipStream_t stream)
{
    (void)n_in; (void)out_size; (void)d_ws; (void)ws_size;
    const float* x = (const float*)d_in[0];
    const float* L = (const float*)d_in[1];
    const float* R = (const float*)d_in[2];
    float* out = (float*)d_out;

    const int batch = in_sizes[0] / N_FULL;   // 8*2048 = 16384 rows
    dim3 grid(batch / TB);                    // 1024 workgroups
    monarch_wmma_f32_kernel<<<grid, NT, 0, stream>>>(x, L, R, out);
}